// GaussianVectorQuantizer_83700322664832
// MI455X (gfx1250) — compile-verified
//
#include <hip/hip_runtime.h>
#include <hip/hip_bf16.h>

// ---------------------------------------------------------------------------
// Problem constants (from reference)
// ---------------------------------------------------------------------------
constexpr int B_    = 64;
constexpr int NPTS  = 1024;
constexpr int DIM   = 256;
constexpr int BOOK  = 2048;
constexpr int NC    = 16;
constexpr int NROWS = B_ * NPTS;           // 65536

// output tuple layout (flat, in return order)
constexpr size_t ZQ_OFF     = 0;
constexpr size_t PREC_OFF   = (size_t)NROWS * DIM;                // 16777216
constexpr size_t LOGITS_OFF = PREC_OFF + 1;                       // 16777217 (4B aligned only!)
constexpr size_t MUMIX_OFF  = LOGITS_OFF + (size_t)NROWS * BOOK;  // 150994945

// workspace layout (floats)
constexpr size_t WS_ZP     = 0;
constexpr size_t WS_RN     = (size_t)NROWS * DIM;   // row norms   [65536]
constexpr size_t WS_RMAX   = WS_RN + NROWS;         // row max     [65536]
constexpr size_t WS_RINV   = WS_RMAX + NROWS;       // 1/row sum   [65536]
constexpr size_t WS_BN     = WS_RINV + NROWS;       // book norms  [2048]

typedef __attribute__((ext_vector_type(2))) float v2f;
typedef __attribute__((ext_vector_type(8))) float v8f;
typedef __attribute__((ext_vector_type(4))) int   v4i;

// address-space-qualified views for the async-LDS builtin
typedef __attribute__((address_space(1))) v4i as1_v4i;   // global
typedef __attribute__((address_space(3))) v4i as3_v4i;   // LDS

// ---------------------------------------------------------------------------
// CDNA5 async global->LDS staging (ASYNCcnt path). Guarded so the file
// compiles on toolchains that don't declare the builtins (register fallback).
// ---------------------------------------------------------------------------
#if __has_builtin(__builtin_amdgcn_global_load_async_to_lds_b128)
#define HAVE_ASYNC_LDS 1
#else
#define HAVE_ASYNC_LDS 0
#endif

__device__ __forceinline__ void copy16_to_lds(const float* g, float* l) {
#if HAVE_ASYNC_LDS
    as1_v4i* gp = (as1_v4i*)const_cast<float*>(g);
    as3_v4i* lp = (as3_v4i*)l;
    __builtin_amdgcn_global_load_async_to_lds_b128(gp, lp, /*imm offset*/0, /*cpol*/0);
#else
    *(float4*)l = *(const float4*)g;
#endif
}

__device__ __forceinline__ void commit_lds_copies() {
#if HAVE_ASYNC_LDS
#if __has_builtin(__builtin_amdgcn_s_wait_asynccnt)
    __builtin_amdgcn_s_wait_asynccnt(0);
#else
    asm volatile("s_wait_asynccnt 0" ::: "memory");
#endif
#endif
}

// deterministic uniform in (0,1) from an index (stand-in for jax PRNG)
__device__ __forceinline__ float u01_hash(unsigned x) {
    x ^= x >> 17; x *= 0xed5ad4bbu;
    x ^= x >> 11; x *= 0xac4c1b51u;
    x ^= x >> 15; x *= 0x31848babu;
    x ^= x >> 14;
    return ((float)(x >> 8) + 0.5f) * (1.0f / 16777216.0f);
}

__device__ __forceinline__ float gumbel_from_idx(unsigned idx) {
    const float EPS = 1e-10f;
    float u = u01_hash(idx);
    return -__logf(-__logf(u + EPS) + EPS);
}

// ---------------------------------------------------------------------------
// K1: mu_mix[b,n,d] = sum_c c_probs[b,c] * mu[c,n,d];  zp = z + mu_mix
// one block per n; thread = d; mu[c,n,d] loaded once and reused over 64 b.
// ---------------------------------------------------------------------------
__global__ __launch_bounds__(256) void k_mumix_zp(
    const float* __restrict__ z, const float* __restrict__ cprobs,
    const float* __restrict__ mu, float* __restrict__ mumix,
    float* __restrict__ zp)
{
    __shared__ float s_cp[B_ * NC];
    const int n = blockIdx.x;
    const int d = threadIdx.x;
    for (int i = threadIdx.x; i < B_ * NC; i += blockDim.x) s_cp[i] = cprobs[i];
    __syncthreads();

    float mur[NC];
#pragma unroll
    for (int c = 0; c < NC; ++c)
        mur[c] = mu[((size_t)c * NPTS + n) * DIM + d];

    for (int b = 0; b < B_; ++b) {
        float mm = 0.0f;
#pragma unroll
        for (int c = 0; c < NC; ++c) mm = fmaf(s_cp[b * NC + c], mur[c], mm);
        const size_t idx = ((size_t)b * NPTS + n) * DIM + d;
        const float zv = z[idx];
        mumix[idx] = mm;       // scalar store (region base is odd-offset)
        zp[idx]    = zv + mm;
    }
}

// ---------------------------------------------------------------------------
// K2: rownorm[r] = sum_d zp[r,d]^2         (one block of 256 per row)
// ---------------------------------------------------------------------------
__global__ __launch_bounds__(256) void k_rownorm(
    const float* __restrict__ zp, float* __restrict__ rn)
{
    __shared__ float red[256];
    const int r = blockIdx.x, t = threadIdx.x;
    float v = zp[(size_t)r * DIM + t];
    red[t] = v * v;
    __syncthreads();
    for (int s = 128; s > 0; s >>= 1) {
        if (t < s) red[t] += red[t + s];
        __syncthreads();
    }
    if (t == 0) rn[r] = red[0];
}

// ---------------------------------------------------------------------------
// K3: booknorm[k] = sum_d book[k,d]^2 ; block 0 also writes precision_q
// ---------------------------------------------------------------------------
__global__ __launch_bounds__(256) void k_booknorm(
    const float* __restrict__ book, const float* __restrict__ lpq,
    float* __restrict__ bn, float* __restrict__ prec_out)
{
    __shared__ float red[256];
    const int r = blockIdx.x, t = threadIdx.x;
    float v = book[(size_t)r * DIM + t];
    red[t] = v * v;
    __syncthreads();
    for (int s = 128; s > 0; s >>= 1) {
        if (t < s) red[t] += red[t + s];
        __syncthreads();
    }
    if (t == 0) bn[r] = red[0];
    if (r == 0 && t == 0) {
        float pq = __expf(lpq[0]);
        prec_out[0] = 0.5f / fmaxf(pq, 1e-10f);
    }
}

// ---------------------------------------------------------------------------
// K4: logits = -(||zp||^2 + ||book||^2 - 2 * zp @ book^T) * prec
// fp32 WMMA GEMM: M=65536, N=2048, K=256. Block tile 128x128, 8 waves in a
// 4(M)x2(N) grid; each wave owns a 32x64 strip = 2x4 v8f accumulators.
// K chunked by 16, tiles staged via async global->LDS.
// ---------------------------------------------------------------------------
#define TM 128
#define TN 128
#define KC 16
#define LDA (KC + 4)   // 20 floats: 16B-aligned rows, conflict-free

__global__ __launch_bounds__(256) void k_logits_gemm(
    const float* __restrict__ zp, const float* __restrict__ book,
    const float* __restrict__ rn, const float* __restrict__ bn,
    const float* __restrict__ lpq, float* __restrict__ logits)
{
    __shared__ __align__(16) float sA[TM][LDA];   // zp tile   [m][k]
    __shared__ __align__(16) float sB[TN][LDA];   // book tile [n][k]  (B = book^T)

    const int t     = threadIdx.x;
    const int lane  = t & 31;
    const int wave  = t >> 5;             // 0..7
    const int wm    = (wave & 3) * 32;    // wave M offset in tile
    const int wn    = (wave >> 2) * 64;   // wave N offset in tile
    const int lm    = lane & 15;          // M (and N) index within 16
    const int lk    = (lane >> 4) << 1;   // K offset: 0 or 2
    const int mBase = blockIdx.x * TM;
    const int nBase = blockIdx.y * TN;

    const v8f zacc = {0.f,0.f,0.f,0.f,0.f,0.f,0.f,0.f};
    v8f acc[2][4] = {{zacc, zacc, zacc, zacc}, {zacc, zacc, zacc, zacc}};

    for (int k0 = 0; k0 < DIM; k0 += KC) {
        // stage A + B tiles: 2 x 2048 floats, 2 x b128 per thread per tile
        const int i0 = t * 8;
#pragma unroll
        for (int h = 0; h < 2; ++h) {
            const int i = i0 + h * 4;
            const int row = i >> 4, kk = i & 15;
            copy16_to_lds(&zp[(size_t)(mBase + row) * DIM + k0 + kk], &sA[row][kk]);
            copy16_to_lds(&book[(size_t)(nBase + row) * DIM + k0 + kk], &sB[row][kk]);
        }
        commit_lds_copies();
        __syncthreads();

#pragma unroll
        for (int kk = 0; kk < KC; kk += 4) {
            v2f a[2];
#pragma unroll
            for (int mi = 0; mi < 2; ++mi) {
                const int row = wm + mi * 16 + lm;
                a[mi].x = sA[row][kk + lk];
                a[mi].y = sA[row][kk + lk + 1];
            }
#pragma unroll
            for (int j = 0; j < 4; ++j) {
                v2f b;
                const int col = wn + j * 16 + lm;
                b.x = sB[col][kk + lk];
                b.y = sB[col][kk + lk + 1];
#pragma unroll
                for (int mi = 0; mi < 2; ++mi)
                    acc[mi][j] = __builtin_amdgcn_wmma_f32_16x16x4_f32(
                        false, a[mi], false, b, (short)0, acc[mi][j], false, false);
            }
        }
        __syncthreads();
    }

    // epilogue: distance -> logits (scalar stores: logits base is odd-offset)
    const float prec = 0.5f / fmaxf(__expf(lpq[0]), 1e-10f);
#pragma unroll
    for (int mi = 0; mi < 2; ++mi) {
        const int mrow0 = mBase + wm + mi * 16 + ((lane >> 4) << 3);
#pragma unroll
        for (int j = 0; j < 4; ++j) {
            const int   n   = nBase + wn + j * 16 + lm;
            const float bnv = bn[n];
#pragma unroll
            for (int r = 0; r < 8; ++r) {
                const int m = mrow0 + r;
                const float s = acc[mi][j][r];
                logits[(size_t)m * BOOK + n] = -(rn[m] + bnv - 2.0f * s) * prec;
            }
        }
    }
}

// ---------------------------------------------------------------------------
// K5: gumbel-softmax stats per row: rowmax, 1/rowsum  of (logits + g)/T
// ---------------------------------------------------------------------------
__global__ __launch_bounds__(256) void k_softmax_stats(
    const float* __restrict__ logits, const int* __restrict__ temp,
    float* __restrict__ rowmax, float* __restrict__ rowinv)
{
    __shared__ float red[256];
    const int r = blockIdx.x, t = threadIdx.x;
    const float invT = 1.0f / (float)temp[0];

    float v[8];
    float mx = -INFINITY;
#pragma unroll
    for (int i = 0; i < 8; ++i) {
        const int c = t + i * 256;
        const float lg = logits[(size_t)r * BOOK + c];
        const float g  = gumbel_from_idx((unsigned)(r * BOOK + c));
        v[i] = (lg + g) * invT;
        mx = fmaxf(mx, v[i]);
    }
    red[t] = mx; __syncthreads();
    for (int s = 128; s > 0; s >>= 1) {
        if (t < s) red[t] = fmaxf(red[t], red[t + s]);
        __syncthreads();
    }
    mx = red[0];
    __syncthreads();

    float sum = 0.0f;
#pragma unroll
    for (int i = 0; i < 8; ++i) sum += __expf(v[i] - mx);
    red[t] = sum; __syncthreads();
    for (int s = 128; s > 0; s >>= 1) {
        if (t < s) red[t] += red[t + s];
        __syncthreads();
    }
    if (t == 0) { rowmax[r] = mx; rowinv[r] = 1.0f / red[0]; }
}

// ---------------------------------------------------------------------------
// K6: zq = softmax(...) @ book.  fp32 WMMA GEMM: M=65536, N=256, K=2048.
// Encodings materialized on the fly into the LDS A-tile (never stored
// globally — saves a 536 MB round trip). B tile staged async.
// ---------------------------------------------------------------------------
#define LDB2 (TN + 4)  // 132 floats: 16B-aligned rows, conflict-free

__global__ __launch_bounds__(256) void k_zq_gemm(
    const float* __restrict__ logits, const float* __restrict__ book,
    const float* __restrict__ rowmax, const float* __restrict__ rowinv,
    const int* __restrict__ temp, float* __restrict__ zq)
{
    __shared__ __align__(16) float sA[TM][LDA];    // encodings tile [m][k]
    __shared__ __align__(16) float sB[KC][LDB2];   // book tile      [k][n]

    const int t     = threadIdx.x;
    const int lane  = t & 31;
    const int wave  = t >> 5;
    const int wm    = (wave & 3) * 32;
    const int wn    = (wave >> 2) * 64;
    const int lm    = lane & 15;
    const int lk    = (lane >> 4) << 1;
    const int mBase = blockIdx.x * TM;
    const int nBase = blockIdx.y * TN;
    const float invT = 1.0f / (float)temp[0];

    const v8f zacc = {0.f,0.f,0.f,0.f,0.f,0.f,0.f,0.f};
    v8f acc[2][4] = {{zacc, zacc, zacc, zacc}, {zacc, zacc, zacc, zacc}};

    for (int k0 = 0; k0 < BOOK; k0 += KC) {
        // stage B: book[k0+kk][nBase+n], 16x128, async 2 x b128 per thread
        const int i0 = t * 8;
#pragma unroll
        for (int h = 0; h < 2; ++h) {
            const int i = i0 + h * 4;
            const int kk = i >> 7, n = i & 127;
            copy16_to_lds(&book[(size_t)(k0 + kk) * DIM + nBase + n], &sB[kk][n]);
        }
        // stage A with softmax transform (scalar loads: logits base odd-offset)
#pragma unroll
        for (int i = t; i < TM * KC; i += 256) {
            const int row = i >> 4, kk = i & 15;
            const int m = mBase + row, k = k0 + kk;
            const float lg = logits[(size_t)m * BOOK + k];
            const float g  = gumbel_from_idx((unsigned)(m * BOOK + k));
            sA[row][kk] = __expf((lg + g) * invT - rowmax[m]) * rowinv[m];
        }
        commit_lds_copies();
        __syncthreads();

#pragma unroll
        for (int kk = 0; kk < KC; kk += 4) {
            v2f a[2];
#pragma unroll
            for (int mi = 0; mi < 2; ++mi) {
                const int row = wm + mi * 16 + lm;
                a[mi].x = sA[row][kk + lk];
                a[mi].y = sA[row][kk + lk + 1];
            }
#pragma unroll
            for (int j = 0; j < 4; ++j) {
                v2f b;
                const int col = wn + j * 16 + lm;
                b.x = sB[kk + lk][col];
                b.y = sB[kk + lk + 1][col];
#pragma unroll
                for (int mi = 0; mi < 2; ++mi)
                    acc[mi][j] = __builtin_amdgcn_wmma_f32_16x16x4_f32(
                        false, a[mi], false, b, (short)0, acc[mi][j], false, false);
            }
        }
        __syncthreads();
    }

#pragma unroll
    for (int mi = 0; mi < 2; ++mi) {
        const int mrow0 = mBase + wm + mi * 16 + ((lane >> 4) << 3);
#pragma unroll
        for (int j = 0; j < 4; ++j) {
            const int n = nBase + wn + j * 16 + lm;
#pragma unroll
            for (int r = 0; r < 8; ++r) {
                const int m = mrow0 + r;
                zq[(size_t)m * DIM + n] = acc[mi][j][r];
            }
        }
    }
}

// ---------------------------------------------------------------------------
// launch
// ---------------------------------------------------------------------------
extern "C" void kernel_launch(void* const* d_in, const int* in_sizes, int n_in,
                              void* d_out, int out_size, void* d_ws, size_t ws_size,
                              hipStream_t stream)
{
    const float* z      = (const float*)d_in[0];
    const float* cprobs = (const float*)d_in[1];
    const float* lpq    = (const float*)d_in[2];
    const float* book   = (const float*)d_in[3];
    const float* mu     = (const float*)d_in[4];
    const int*   temp   = (const int*)  d_in[5];
    // d_in[6] (is_train) unused by the math

    float* out      = (float*)d_out;
    float* zq       = out + ZQ_OFF;
    float* prec_out = out + PREC_OFF;
    float* logits   = out + LOGITS_OFF;
    float* mumix    = out + MUMIX_OFF;

    float* ws     = (float*)d_ws;
    float* zp     = ws + WS_ZP;
    float* rn     = ws + WS_RN;
    float* rowmax = ws + WS_RMAX;
    float* rowinv = ws + WS_RINV;
    float* bn     = ws + WS_BN;

    k_mumix_zp     <<<NPTS, 256, 0, stream>>>(z, cprobs, mu, mumix, zp);
    k_rownorm      <<<NROWS, 256, 0, stream>>>(zp, rn);
    k_booknorm     <<<BOOK, 256, 0, stream>>>(book, lpq, bn, prec_out);
    k_logits_gemm  <<<dim3(NROWS / TM, BOOK / TN), 256, 0, stream>>>(zp, book, rn, bn, lpq, logits);
    k_softmax_stats<<<NROWS, 256, 0, stream>>>(logits, temp, rowmax, rowinv);
    k_zq_gemm      <<<dim3(NROWS / TM, DIM / TN), 256, 0, stream>>>(logits, book, rowmax, rowinv, temp, zq);
}